// BaseRNN_73821897884188
// MI455X (gfx1250) — compile-verified
//
#include <hip/hip_runtime.h>
#include <math.h>

typedef float v2f __attribute__((ext_vector_type(2)));
typedef float v8f __attribute__((ext_vector_type(8)));

#define ALPHA 0.5f

// ---- WMMA f32 16x16x4 fragment layout (wave32) ----
// A (16x4): lanes 0-15 -> M=lane, VGPR0/1 = K {0,1}; lanes 16-31 -> K {2,3}
// B (4x16): lanes 0-15 -> N=lane, VGPR0/1 = K {0,1}; lanes 16-31 -> K {2,3}
// C/D (16x16): VGPR v, lanes 0-15 -> M=v, N=lane; lanes 16-31 -> M=v+8

static __device__ __forceinline__ v8f wmma4(v2f a, v2f b, v8f c) {
  return __builtin_amdgcn_wmma_f32_16x16x4_f32(false, a, false, b, (short)0, c,
                                               false, false);
}
static __device__ __forceinline__ v2f ldp2(const float* p) {  // contiguous K pair
  v2f r; r.x = p[0]; r.y = p[1]; return r;
}
static __device__ __forceinline__ v2f ldps(const float* p, int stride) {  // strided K pair
  v2f r; r.x = p[0]; r.y = p[stride]; return r;
}
static __device__ __forceinline__ void st_c(float* C, int ldc, int row0, int col0, v8f c) {
  int lane = threadIdx.x & 31;
  int n = col0 + (lane & 15);
  int mbase = row0 + ((lane >> 4) << 3);
#pragma unroll
  for (int v = 0; v < 8; ++v) C[(size_t)(mbase + v) * ldc + n] = c[v];
}
// generic A-frag from row-major (used only in short LDS loop)
static __device__ __forceinline__ v2f ld_a(const float* A, int lda, int row0, int kb) {
  int lane = threadIdx.x & 31;
  int m = lane & 15;
  int k = kb + ((lane >> 4) << 1);
  const float* p = A + (size_t)(row0 + m) * lda + k;
  v2f a; a.x = p[0]; a.y = p[1]; return a;
}

// ---- init: scatter h0 (B,H) into hidden_states[:,0,:] ----
__global__ void k_init_h(const float* __restrict__ h0, float* __restrict__ hs,
                         int B, int T, int H) {
  int i = blockIdx.x * blockDim.x + threadIdx.x;
  if (i < B * H) {
    int b = i / H, j = i % H;
    hs[(size_t)b * (T + 1) * H + j] = h0[i];
  }
}

// ---- step 1: pre = h@Wrec + x_t@Win ; h_new = (1-a)h + a*tanh(pre)
//      one wave per 32x16 block (both batch tiles share the B fragment)
__global__ void k_step_h(const float* __restrict__ x, const float* __restrict__ Wrec,
                         const float* __restrict__ Win, float* __restrict__ hs,
                         int B, int T, int I, int H, int t) {
  int col0 = blockIdx.x << 4;
  int ldh = (T + 1) * H;
  int lane = threadIdx.x & 31;
  int mn = lane & 15;
  int kof = (lane >> 4) << 1;  // 0 or 2
  // A = h_prev : A[b][k] = hs[b*ldh + t*H + k]
  const float* pa0 = hs + (size_t)mn * ldh + (size_t)t * H + kof;
  const float* pa1 = pa0 + (size_t)16 * ldh;
  const float* pb  = Wrec + (size_t)kof * H + col0 + mn;
  v8f acc0 = {}, acc1 = {};
  for (int kb = 0; kb < H; kb += 4) {
    v2f a0 = ldp2(pa0), a1 = ldp2(pa1);
    v2f b = ldps(pb, H);
    acc0 = wmma4(a0, b, acc0);
    acc1 = wmma4(a1, b, acc1);
    pa0 += 4; pa1 += 4; pb += (size_t)4 * H;
  }
  // A = x_t : A[b][k] = x[b*(T*I) + t*I + k]
  const float* px0 = x + (size_t)mn * T * I + (size_t)t * I + kof;
  const float* px1 = px0 + (size_t)16 * T * I;
  const float* pw  = Win + (size_t)kof * H + col0 + mn;
  for (int kb = 0; kb < I; kb += 4) {
    v2f a0 = ldp2(px0), a1 = ldp2(px1);
    v2f b = ldps(pw, H);
    acc0 = wmma4(a0, b, acc0);
    acc1 = wmma4(a1, b, acc1);
    px0 += 4; px1 += 4; pw += (size_t)4 * H;
  }
  int n = col0 + mn;
  int mb = (lane >> 4) << 3;
#pragma unroll
  for (int v = 0; v < 8; ++v) {
    int b0 = mb + v, b1 = b0 + 16;
    size_t o0 = ((size_t)b0 * (T + 1) + t) * H + n;
    size_t o1 = ((size_t)b1 * (T + 1) + t) * H + n;
    hs[o0 + H] = (1.f - ALPHA) * hs[o0] + ALPHA * tanhf(acc0[v]);
    hs[o1 + H] = (1.f - ALPHA) * hs[o1] + ALPHA * tanhf(acc1[v]);
  }
}

// ---- step 2 (fused): Ph = P@h^T (16x32/wave), hP = h@P (32x16/wave),
//      out = h@Wout, e = out - tgt ----
__global__ void k_misc(const float* __restrict__ hs, const float* __restrict__ Pin,
                       const float* __restrict__ Wout, const float* __restrict__ tgt,
                       float* __restrict__ outs, float* __restrict__ Ph,
                       float* __restrict__ hP, float* __restrict__ ebuf,
                       int B, int T, int H, int O, int t) {
  int ldh = (T + 1) * H;
  const float* hc = hs + (size_t)(t + 1) * H;  // current (post-update) h
  int nbt = B >> 4, nht = H >> 4;
  int lane = threadIdx.x & 31;
  int mn = lane & 15;
  int kof = (lane >> 4) << 1;
  int id = blockIdx.x;
  if (id < nht) {                              // Ph (H x B): 16 rows x 32 batch
    int row0 = id << 4;
    // A = Pin rows
    const float* pa = Pin + (size_t)(row0 + mn) * H + kof;
    // B = h^T : B[k][b] = hc[b*ldh + k]  (contiguous in k)
    const float* pb0 = hc + (size_t)mn * ldh + kof;
    const float* pb1 = pb0 + (size_t)16 * ldh;
    v8f acc0 = {}, acc1 = {};
    for (int kb = 0; kb < H; kb += 4) {
      v2f a = ldp2(pa);
      v2f b0 = ldp2(pb0), b1 = ldp2(pb1);
      acc0 = wmma4(a, b0, acc0);
      acc1 = wmma4(a, b1, acc1);
      pa += 4; pb0 += 4; pb1 += 4;
    }
    st_c(Ph, B, row0, 0, acc0);
    st_c(Ph, B, row0, 16, acc1);
    return;
  }
  id -= nht;
  if (id < nht) {                              // hP (B x H): 32 batch x 16 cols
    int col0 = id << 4;
    const float* pa0 = hc + (size_t)mn * ldh + kof;   // A = h rows (batch)
    const float* pa1 = pa0 + (size_t)16 * ldh;
    const float* pb  = Pin + (size_t)kof * H + col0 + mn;
    v8f acc0 = {}, acc1 = {};
    for (int kb = 0; kb < H; kb += 4) {
      v2f a0 = ldp2(pa0), a1 = ldp2(pa1);
      v2f b = ldps(pb, H);
      acc0 = wmma4(a0, b, acc0);
      acc1 = wmma4(a1, b, acc1);
      pa0 += 4; pa1 += 4; pb += (size_t)4 * H;
    }
    st_c(hP, H, 0, col0, acc0);
    st_c(hP, H, 16, col0, acc1);
    return;
  }
  id -= nht;                                   // out/e (B x O), 16x16 tiles
  int mt = id % nbt, nt = id / nbt;
  int row0 = mt << 4, col0 = nt << 4;
  const float* pa = hc + (size_t)(row0 + mn) * ldh + kof;
  const float* pb = Wout + (size_t)kof * O + col0 + mn;
  v8f acc = {};
  for (int kb = 0; kb < H; kb += 4) {
    acc = wmma4(ldp2(pa), ldps(pb, O), acc);
    pa += 4; pb += (size_t)4 * O;
  }
  int n = col0 + mn;
  int mb = row0 + ((lane >> 4) << 3);
#pragma unroll
  for (int v = 0; v < 8; ++v) {
    int b = mb + v;
    float o = acc[v];
    outs[((size_t)b * T + t) * O + n] = o;
    ebuf[b * O + n] = o - tgt[((size_t)b * T + t) * O + n];
  }
}

// ---- step 3: G = Ph @ h  (H x H, K = B), 32x32 block per wave ----
__global__ void k_G(const float* __restrict__ Ph, const float* __restrict__ hs,
                    float* __restrict__ G, int B, int T, int H, int t) {
  int n2 = H >> 5;
  int it = blockIdx.x / n2, jt = blockIdx.x % n2;
  int row0 = it << 5, col0 = jt << 5;
  int ldh = (T + 1) * H;
  const float* hc = hs + (size_t)(t + 1) * H;
  int lane = threadIdx.x & 31;
  int mn = lane & 15;
  int kof = (lane >> 4) << 1;
  const float* pa0 = Ph + (size_t)(row0 + mn) * B + kof;
  const float* pa1 = pa0 + (size_t)16 * B;
  const float* pb0 = hc + (size_t)kof * ldh + col0 + mn;
  const float* pb1 = pb0 + 16;
  v8f c00 = {}, c01 = {}, c10 = {}, c11 = {};
  for (int kb = 0; kb < B; kb += 4) {
    v2f a0 = ldp2(pa0), a1 = ldp2(pa1);
    v2f b0 = ldps(pb0, ldh), b1 = ldps(pb1, ldh);
    c00 = wmma4(a0, b0, c00); c01 = wmma4(a0, b1, c01);
    c10 = wmma4(a1, b0, c10); c11 = wmma4(a1, b1, c11);
    pa0 += 4; pa1 += 4;
    pb0 += (size_t)4 * ldh; pb1 += (size_t)4 * ldh;
  }
  st_c(G, H, row0,      col0,      c00);
  st_c(G, H, row0,      col0 + 16, c01);
  st_c(G, H, row0 + 16, col0,      c10);
  st_c(G, H, row0 + 16, col0 + 16, c11);
}

// ---- step 4: P_out = P_in - (G@P_in) / (1 + h^T @ hP) ; 32x32 block per wave ----
__global__ void k_Pupd(const float* __restrict__ G, const float* __restrict__ Pin,
                       const float* __restrict__ hs, const float* __restrict__ hP,
                       float* __restrict__ Pout, int B, int T, int H, int t) {
  int n2 = H >> 5;
  int it = blockIdx.x / n2, jt = blockIdx.x % n2;
  int row0 = it << 5, col0 = jt << 5;
  int ldh = (T + 1) * H;
  const float* hc = hs + (size_t)(t + 1) * H;
  int lane = threadIdx.x & 31;
  int mn = lane & 15;
  int kof = (lane >> 4) << 1;

  // U = G @ Pin over K = H
  const float* pa0 = G + (size_t)(row0 + mn) * H + kof;
  const float* pa1 = pa0 + (size_t)16 * H;
  const float* pb0 = Pin + (size_t)kof * H + col0 + mn;
  const float* pb1 = pb0 + 16;
  v8f u00 = {}, u01 = {}, u10 = {}, u11 = {};
  for (int kb = 0; kb < H; kb += 4) {
    v2f a0 = ldp2(pa0), a1 = ldp2(pa1);
    v2f b0 = ldps(pb0, H), b1 = ldps(pb1, H);
    u00 = wmma4(a0, b0, u00); u01 = wmma4(a0, b1, u01);
    u10 = wmma4(a1, b0, u10); u11 = wmma4(a1, b1, u11);
    pa0 += 4; pa1 += 4;
    pb0 += (size_t)4 * H; pb1 += (size_t)4 * H;
  }

  // bottom = h^T @ hP over K = B   (A[m][k] = hc[k*ldh + row0+m])
  const float* pc0 = hc + (size_t)kof * ldh + row0 + mn;
  const float* pc1 = pc0 + 16;
  const float* pd0 = hP + (size_t)kof * H + col0 + mn;
  const float* pd1 = pd0 + 16;
  v8f t00 = {}, t01 = {}, t10 = {}, t11 = {};
  for (int kb = 0; kb < B; kb += 4) {
    v2f a0 = ldps(pc0, ldh), a1 = ldps(pc1, ldh);
    v2f b0 = ldps(pd0, H), b1 = ldps(pd1, H);
    t00 = wmma4(a0, b0, t00); t01 = wmma4(a0, b1, t01);
    t10 = wmma4(a1, b0, t10); t11 = wmma4(a1, b1, t11);
    pc0 += (size_t)4 * ldh; pc1 += (size_t)4 * ldh;
    pd0 += (size_t)4 * H;   pd1 += (size_t)4 * H;
  }

  int mb = (lane >> 4) << 3;
#pragma unroll
  for (int v = 0; v < 8; ++v) {
    size_t r0 = (size_t)(row0 + mb + v) * H;
    size_t r1 = (size_t)(row0 + 16 + mb + v) * H;
    size_t i00 = r0 + col0 + mn, i01 = r0 + col0 + 16 + mn;
    size_t i10 = r1 + col0 + mn, i11 = r1 + col0 + 16 + mn;
    Pout[i00] = Pin[i00] - u00[v] / (1.f + t00[v]);
    Pout[i01] = Pin[i01] - u01[v] / (1.f + t01[v]);
    Pout[i10] = Pin[i10] - u10[v] / (1.f + t10[v]);
    Pout[i11] = Pin[i11] - u11[v] / (1.f + t11[v]);
  }
}

// ---- step 5: Wout -= (P_new @ h^T) @ e ; Ph2 strip staged through LDS ----
__global__ void k_Wupd(const float* __restrict__ Pnew, const float* __restrict__ hs,
                       const float* __restrict__ ebuf, float* __restrict__ Wout,
                       int B, int T, int H, int O, int t) {
  __shared__ float sm[16 * 32];  // 16 x B strip of P_new @ h^T (B == 32)
  int noT = O >> 4;
  int it = blockIdx.x / noT, jt = blockIdx.x % noT;
  int row0 = it << 4, col0 = jt << 4;
  int ldh = (T + 1) * H;
  const float* hc = hs + (size_t)(t + 1) * H;
  int lane = threadIdx.x & 31;
  int mn = lane & 15;
  int kof = (lane >> 4) << 1;
  const float* pa  = Pnew + (size_t)(row0 + mn) * H + kof;
  const float* pb0 = hc + (size_t)mn * ldh + kof;     // h^T, contiguous in k
  const float* pb1 = pb0 + (size_t)16 * ldh;
  v8f acc0 = {}, acc1 = {};
  for (int kb = 0; kb < H; kb += 4) {
    v2f a = ldp2(pa);
    acc0 = wmma4(a, ldp2(pb0), acc0);
    acc1 = wmma4(a, ldp2(pb1), acc1);
    pa += 4; pb0 += 4; pb1 += 4;
  }
  st_c(sm, B, 0, 0, acc0);
  st_c(sm, B, 0, 16, acc1);
  __syncthreads();
  const float* pe = ebuf + (size_t)kof * O + col0 + mn;
  v8f w = {};
  for (int kb = 0; kb < B; kb += 4) {
    w = wmma4(ld_a(sm, B, 0, kb), ldps(pe, O), w);
    pe += (size_t)4 * O;
  }
  int mb = row0 + ((lane >> 4) << 3);
#pragma unroll
  for (int v = 0; v < 8; ++v) {
    size_t idx = (size_t)(mb + v) * O + col0 + mn;
    Wout[idx] -= w[v];
  }
}

extern "C" void kernel_launch(void* const* d_in, const int* in_sizes, int n_in,
                              void* d_out, int out_size, void* d_ws, size_t ws_size,
                              hipStream_t stream) {
  const float* x     = (const float*)d_in[0];
  const float* h0    = (const float*)d_in[1];
  const float* P0i   = (const float*)d_in[2];
  const float* tgt   = (const float*)d_in[3];
  const float* Wrec  = (const float*)d_in[4];
  const float* Win   = (const float*)d_in[5];
  const float* Wout0 = (const float*)d_in[6];

  int H = (int)(sqrt((double)in_sizes[2]) + 0.5);
  int B = in_sizes[1] / H;
  int O = in_sizes[6] / H;
  int I = in_sizes[5] / H;
  int T = in_sizes[0] / (B * I);

  float* outs = (float*)d_out;                       // (B,T,O)
  float* hs   = outs + (size_t)B * T * O;            // (B,T+1,H)
  float* Pfin = hs + (size_t)B * (T + 1) * H;        // (H,H)

  float* w = (float*)d_ws;
  float* Pb0 = w;
  float* Pb1 = Pb0 + (size_t)H * H;
  float* G   = Pb1 + (size_t)H * H;
  float* Ph  = G + (size_t)H * H;
  float* hP  = Ph + (size_t)H * B;
  float* eb  = hP + (size_t)B * H;
  float* Wo  = eb + (size_t)B * O;
  float* Pb[2] = {Pb0, Pb1};

  hipMemcpyAsync(Pb0, P0i, (size_t)H * H * sizeof(float), hipMemcpyDeviceToDevice, stream);
  hipMemcpyAsync(Wo, Wout0, (size_t)H * O * sizeof(float), hipMemcpyDeviceToDevice, stream);
  k_init_h<<<(B * H + 255) / 256, 256, 0, stream>>>(h0, hs, B, T, H);

  int nbt = B >> 4, nht = H >> 4, noT = O >> 4, n2 = H >> 5;
  for (int t = 0; t < T; ++t) {
    const float* Pin = Pb[t & 1];
    float* Pnext     = Pb[(t + 1) & 1];
    k_step_h<<<nht, 32, 0, stream>>>(x, Wrec, Win, hs, B, T, I, H, t);
    k_misc<<<nht + nht + nbt * noT, 32, 0, stream>>>(
        hs, Pin, Wo, tgt, outs, Ph, hP, eb, B, T, H, O, t);
    k_G<<<n2 * n2, 32, 0, stream>>>(Ph, hs, G, B, T, H, t);
    k_Pupd<<<n2 * n2, 32, 0, stream>>>(G, Pin, hs, hP, Pnext, B, T, H, t);
    k_Wupd<<<nht * noT, 32, 0, stream>>>(Pnext, hs, eb, Wo, B, T, H, O, t);
  }
  hipMemcpyAsync(Pfin, Pb[T & 1], (size_t)H * H * sizeof(float), hipMemcpyDeviceToDevice, stream);
}